// GATEdgeAttr_27908697489550
// MI455X (gfx1250) — compile-verified
//
#include <hip/hip_runtime.h>
#include <math.h>

typedef __attribute__((ext_vector_type(2))) float v2f;
typedef __attribute__((ext_vector_type(8))) float v8f;

#define NEG_SLOPE 0.2f

// ---------------------------------------------------------------- utilities
__device__ __forceinline__ void atomic_max_f32(float* addr, float val) {
  // ordered-int trick: works for mixed-sign floats
  if (val >= 0.0f)
    atomicMax((int*)addr, __float_as_int(val));
  else
    atomicMin((unsigned int*)addr, (unsigned int)__float_as_int(val));
}

__global__ void k_fill_f32(float* __restrict__ p, float v, long long n) {
  long long t = (long long)blockIdx.x * blockDim.x + threadIdx.x;
  if (t < n) p[t] = v;
}

// ------------------------------------------------- self loops / merged edges
__global__ void k_edge_prepare(const int* __restrict__ src, const int* __restrict__ dst,
                               const float* __restrict__ ea,
                               int* __restrict__ srcM, int* __restrict__ dstM,
                               float* __restrict__ eaM,
                               float* __restrict__ sumA, float* __restrict__ cntA, int E) {
  int e = blockIdx.x * blockDim.x + threadIdx.x;
  if (e >= E) return;
  int s = src[e], d = dst[e];
  float a = ea[e];
  srcM[e] = s;
  dstM[e] = d;
  eaM[e] = a;
  atomicAdd(&sumA[d], a);
  atomicAdd(&cntA[d], 1.0f);
}

__global__ void k_self_loops(int* __restrict__ srcM, int* __restrict__ dstM,
                             float* __restrict__ eaM,
                             const float* __restrict__ sumA, const float* __restrict__ cntA,
                             int E, int N) {
  int n = blockIdx.x * blockDim.x + threadIdx.x;
  if (n >= N) return;
  srcM[E + n] = n;
  dstM[E + n] = n;
  eaM[E + n] = sumA[n] / fmaxf(cntA[n], 1.0f);  // fill_value='mean'
}

// ------------------------------------------------------------- fp32 WMMA GEMM
// Y[N x 128] = X[N x K] @ W[K x 128], exact fp32 via V_WMMA_F32_16X16X4_F32.
// 256 threads = 8 waves per block; each wave owns one 16-row tile and all
// 8 column tiles. W staged in LDS in 64x128 chunks (32 KB) using the CDNA5
// async-to-LDS path (GLOBAL_LOAD_ASYNC_TO_LDS_B128, tracked by ASYNCcnt):
// bulk copy bypasses VGPRs entirely.
template <int K>
__global__ __launch_bounds__(256) void k_wmma_gemm(const float* __restrict__ X,
                                                   const float* __restrict__ W,
                                                   float* __restrict__ Y, int nrows) {
  __shared__ float lw[64 * 128];  // single LDS alloc -> byte offset 0
  const int tid = threadIdx.x;
  const int wave = tid >> 5;
  const int lane = tid & 31;
  const int lhalf = lane >> 4;  // 0: K=0,1  1: K=2,3 (ISA 16x4 f32 A layout)
  const int lm = lane & 15;
  const int row0 = (blockIdx.x * 8 + wave) * 16;
  const bool active = (row0 < nrows);  // wave-uniform -> EXEC all-1s for WMMA

  v8f acc[8];
#pragma unroll
  for (int i = 0; i < 8; ++i)
#pragma unroll
    for (int j = 0; j < 8; ++j) acc[i][j] = 0.0f;

  int arow_i = row0 + lm;
  if (arow_i >= nrows) arow_i = nrows - 1;  // safety clamp (N%16==0 normally)

  for (int kc = 0; kc < K; kc += 64) {
    // ---- async stage W[kc..kc+63][0..127] -> LDS (row-major), 16B per lane
    {
      const char* gw = (const char*)(W + (size_t)kc * 128);
#pragma unroll
      for (int j = 0; j < 8; ++j) {
        unsigned ldsOff = (unsigned)((tid + j * 256) * 16);
        asm volatile("global_load_async_to_lds_b128 %0, %1, off"
                     :
                     : "v"(ldsOff), "v"(gw + ldsOff)
                     : "memory");
      }
      asm volatile("s_wait_asynccnt 0x0" ::: "memory");
    }
    __syncthreads();

    if (active) {
      const float* arow = X + (size_t)arow_i * K + kc;
      if (kc + 64 < K) __builtin_prefetch(arow + 64, 0, 1);  // next A chunk
#pragma unroll
      for (int kk = 0; kk < 64; kk += 4) {
        v2f a;
        a[0] = arow[kk + lhalf * 2 + 0];
        a[1] = arow[kk + lhalf * 2 + 1];
#pragma unroll
        for (int nt = 0; nt < 8; ++nt) {
          v2f b;  // B 4x16 f32: lanes 0-15 K=0,1 ; lanes 16-31 K=2,3 ; N=lane&15
          b[0] = lw[(kk + lhalf * 2 + 0) * 128 + nt * 16 + lm];
          b[1] = lw[(kk + lhalf * 2 + 1) * 128 + nt * 16 + lm];
          acc[nt] = __builtin_amdgcn_wmma_f32_16x16x4_f32(
              false, a, false, b, (short)0, acc[nt], false, false);
        }
      }
    }
    __syncthreads();
  }

  if (!active) return;
#pragma unroll
  for (int nt = 0; nt < 8; ++nt)
#pragma unroll
    for (int r = 0; r < 8; ++r) {
      int row = row0 + r + lhalf * 8;  // C/D layout: vgpr r -> M=r / M=r+8
      if (row < nrows) Y[(size_t)row * 128 + nt * 16 + lm] = acc[nt][r];
    }
}

// ------------------------------------------------------------ small helpers
// wedot[h] = sum_c We[h*C+c] * ae[h*C+c]   (since e = ea * We is rank-1)
__global__ void k_wedot(const float* __restrict__ We, const float* __restrict__ aeP,
                        float* __restrict__ out, int H, int C) {
  int h = threadIdx.x;
  if (h >= H) return;
  float s = 0.0f;
  for (int c = 0; c < C; ++c) s += We[h * C + c] * aeP[h * C + c];
  out[h] = s;
}

__global__ void k_node_logits(const float* __restrict__ hb, const float* __restrict__ aS,
                              const float* __restrict__ aD, float* __restrict__ ls,
                              float* __restrict__ ld_, int N, int H, int C) {
  long long t = (long long)blockIdx.x * blockDim.x + threadIdx.x;
  if (t >= (long long)N * H) return;
  int n = (int)(t / H), h = (int)(t % H);
  const float* hp = hb + (size_t)n * H * C + (size_t)h * C;
  float s = 0.0f, d = 0.0f;
  for (int c = 0; c < C; ++c) {
    float v = hp[c];
    s += v * aS[h * C + c];
    d += v * aD[h * C + c];
  }
  ls[t] = s;
  ld_[t] = d;
}

__global__ void k_edge_logits(const int* __restrict__ srcM, const int* __restrict__ dstM,
                              const float* __restrict__ eaM, const float* __restrict__ ls,
                              const float* __restrict__ ld_, const float* __restrict__ wedot,
                              float* __restrict__ elog, float* __restrict__ mx,
                              long long EH, int H) {
  long long t = (long long)blockIdx.x * blockDim.x + threadIdx.x;
  if (t >= EH) return;
  long long e = t / H;
  int h = (int)(t % H);
  int s = srcM[e], d = dstM[e];
  float l = ls[(size_t)s * H + h] + ld_[(size_t)d * H + h] + eaM[e] * wedot[h];
  l = (l > 0.0f) ? l : NEG_SLOPE * l;
  elog[t] = l;
  atomic_max_f32(&mx[(size_t)d * H + h], l);
}

__global__ void k_edge_expsum(const int* __restrict__ dstM, float* __restrict__ elog,
                              const float* __restrict__ mx, float* __restrict__ denom,
                              long long EH, int H) {
  long long t = (long long)blockIdx.x * blockDim.x + threadIdx.x;
  if (t >= EH) return;
  long long e = t / H;
  int h = (int)(t % H);
  int d = dstM[e];
  float ex = __expf(elog[t] - mx[(size_t)d * H + h]);
  elog[t] = ex;
  atomicAdd(&denom[(size_t)d * H + h], ex);
}

__global__ void k_aggregate(const int* __restrict__ srcM, const int* __restrict__ dstM,
                            const float* __restrict__ elog, const float* __restrict__ denom,
                            const float* __restrict__ hb, float* __restrict__ agg,
                            long long EF, int H, int C, int HC) {
  long long t = (long long)blockIdx.x * blockDim.x + threadIdx.x;
  if (t >= EF) return;
  long long e = t / HC;
  int f = (int)(t % HC);
  int h = f / C;
  int s = srcM[e], d = dstM[e];
  float alpha = elog[e * H + h] / (denom[(size_t)d * H + h] + 1e-16f);
  atomicAdd(&agg[(size_t)d * HC + f], alpha * hb[(size_t)s * HC + f]);
}

__global__ void k_bias_act(float* __restrict__ io, const float* __restrict__ b,
                           long long NF, int HC, int relu) {
  long long t = (long long)blockIdx.x * blockDim.x + threadIdx.x;
  if (t >= NF) return;
  float v = io[t] + b[(int)(t % HC)];
  io[t] = relu ? fmaxf(v, 0.0f) : v;
}

// layer-3 GEMV: h3[n] = h2[n,:] . W3
__global__ void k_dot128(const float* __restrict__ X, const float* __restrict__ W3,
                         float* __restrict__ out, int N) {
  int n = blockIdx.x * blockDim.x + threadIdx.x;
  if (n >= N) return;
  const float* xp = X + (size_t)n * 128;
  float s = 0.0f;
#pragma unroll
  for (int f = 0; f < 128; ++f) s += xp[f] * W3[f];
  out[n] = s;
}

// ---------------------------------------------------------------- launcher
extern "C" void kernel_launch(void* const* d_in, const int* in_sizes, int n_in,
                              void* d_out, int out_size, void* d_ws, size_t ws_size,
                              hipStream_t stream) {
  const float* x   = (const float*)d_in[0];
  const int*   ei  = (const int*)d_in[1];
  const float* ea  = (const float*)d_in[2];
  const float* W1  = (const float*)d_in[3];
  const float* We1 = (const float*)d_in[4];
  const float* as1 = (const float*)d_in[5];
  const float* ad1 = (const float*)d_in[6];
  const float* ae1 = (const float*)d_in[7];
  const float* b1  = (const float*)d_in[8];
  const float* W2  = (const float*)d_in[9];
  const float* We2 = (const float*)d_in[10];
  const float* as2 = (const float*)d_in[11];
  const float* ad2 = (const float*)d_in[12];
  const float* ae2 = (const float*)d_in[13];
  const float* b2  = (const float*)d_in[14];
  const float* W3  = (const float*)d_in[15];
  const float* We3 = (const float*)d_in[16];
  const float* as3 = (const float*)d_in[17];
  const float* ad3 = (const float*)d_in[18];
  const float* ae3 = (const float*)d_in[19];
  const float* b3  = (const float*)d_in[20];
  float* out = (float*)d_out;
  (void)n_in; (void)out_size; (void)ws_size;

  const int N = in_sizes[0] / 256;  // F_IN = 256
  const int E = in_sizes[2];        // edge_attr is [E,1]
  const int Etot = E + N;
  const int H = 4, C = 32, HC = 128;

  // carve workspace (256B aligned chunks)
  char* wp = (char*)d_ws;
  auto carve = [&](size_t bytes) -> char* {
    char* p = wp;
    wp += (bytes + 255) & ~(size_t)255;
    return p;
  };
  int*   srcM = (int*)carve((size_t)Etot * 4);
  int*   dstM = (int*)carve((size_t)Etot * 4);
  float* eaM  = (float*)carve((size_t)Etot * 4);
  float* sumA = (float*)carve((size_t)N * 4);
  float* cntA = (float*)carve((size_t)N * 4);
  float* bufA = (float*)carve((size_t)N * HC * 4);  // GEMM output / node feats
  float* bufB = (float*)carve((size_t)N * HC * 4);  // aggregation / next input
  float* elog = (float*)carve((size_t)Etot * H * 4);
  float* ls   = (float*)carve((size_t)N * H * 4);
  float* ld_  = (float*)carve((size_t)N * H * 4);
  float* mx   = (float*)carve((size_t)N * H * 4);
  float* den  = (float*)carve((size_t)N * H * 4);
  float* wed  = (float*)carve(256);
  float* h3   = (float*)carve((size_t)N * 4);

  const int T = 256;
  auto nb = [](long long n, int t) { return (unsigned)((n + t - 1) / t); };
  const int mtiles = (N + 15) / 16;
  const unsigned gemmBlocks = (unsigned)((mtiles + 7) / 8);

  // ---- self loops + merged edge arrays
  k_fill_f32<<<nb(N, T), T, 0, stream>>>(sumA, 0.0f, N);
  k_fill_f32<<<nb(N, T), T, 0, stream>>>(cntA, 0.0f, N);
  k_edge_prepare<<<nb(E, T), T, 0, stream>>>(ei, ei + E, ea, srcM, dstM, eaM, sumA, cntA, E);
  k_self_loops<<<nb(N, T), T, 0, stream>>>(srcM, dstM, eaM, sumA, cntA, E, N);

  // ================= layer 1: 256 -> 4x32 =================
  k_wmma_gemm<256><<<gemmBlocks, T, 0, stream>>>(x, W1, bufA, N);
  k_wedot<<<1, 32, 0, stream>>>(We1, ae1, wed, H, C);
  k_node_logits<<<nb((long long)N * H, T), T, 0, stream>>>(bufA, as1, ad1, ls, ld_, N, H, C);
  k_fill_f32<<<nb((long long)N * H, T), T, 0, stream>>>(mx, -1e30f, (long long)N * H);
  k_fill_f32<<<nb((long long)N * H, T), T, 0, stream>>>(den, 0.0f, (long long)N * H);
  k_fill_f32<<<nb((long long)N * HC, T), T, 0, stream>>>(bufB, 0.0f, (long long)N * HC);
  k_edge_logits<<<nb((long long)Etot * H, T), T, 0, stream>>>(srcM, dstM, eaM, ls, ld_, wed,
                                                              elog, mx, (long long)Etot * H, H);
  k_edge_expsum<<<nb((long long)Etot * H, T), T, 0, stream>>>(dstM, elog, mx, den,
                                                              (long long)Etot * H, H);
  k_aggregate<<<nb((long long)Etot * HC, T), T, 0, stream>>>(srcM, dstM, elog, den, bufA, bufB,
                                                             (long long)Etot * HC, H, C, HC);
  k_bias_act<<<nb((long long)N * HC, T), T, 0, stream>>>(bufB, b1, (long long)N * HC, HC, 1);

  // ================= layer 2: 128 -> 4x32 =================
  k_wmma_gemm<128><<<gemmBlocks, T, 0, stream>>>(bufB, W2, bufA, N);
  k_wedot<<<1, 32, 0, stream>>>(We2, ae2, wed, H, C);
  k_node_logits<<<nb((long long)N * H, T), T, 0, stream>>>(bufA, as2, ad2, ls, ld_, N, H, C);
  k_fill_f32<<<nb((long long)N * H, T), T, 0, stream>>>(mx, -1e30f, (long long)N * H);
  k_fill_f32<<<nb((long long)N * H, T), T, 0, stream>>>(den, 0.0f, (long long)N * H);
  k_fill_f32<<<nb((long long)N * HC, T), T, 0, stream>>>(bufB, 0.0f, (long long)N * HC);
  k_edge_logits<<<nb((long long)Etot * H, T), T, 0, stream>>>(srcM, dstM, eaM, ls, ld_, wed,
                                                              elog, mx, (long long)Etot * H, H);
  k_edge_expsum<<<nb((long long)Etot * H, T), T, 0, stream>>>(dstM, elog, mx, den,
                                                              (long long)Etot * H, H);
  k_aggregate<<<nb((long long)Etot * HC, T), T, 0, stream>>>(srcM, dstM, elog, den, bufA, bufB,
                                                             (long long)Etot * HC, H, C, HC);
  k_bias_act<<<nb((long long)N * HC, T), T, 0, stream>>>(bufB, b2, (long long)N * HC, HC, 1);

  // ================= layer 3: 128 -> 1x1 =================
  k_dot128<<<nb(N, T), T, 0, stream>>>(bufB, W3, h3, N);
  k_wedot<<<1, 32, 0, stream>>>(We3, ae3, wed, 1, 1);
  k_node_logits<<<nb(N, T), T, 0, stream>>>(h3, as3, ad3, ls, ld_, N, 1, 1);
  k_fill_f32<<<nb(N, T), T, 0, stream>>>(mx, -1e30f, N);
  k_fill_f32<<<nb(N, T), T, 0, stream>>>(den, 0.0f, N);
  k_fill_f32<<<nb(N, T), T, 0, stream>>>(out, 0.0f, N);
  k_edge_logits<<<nb(Etot, T), T, 0, stream>>>(srcM, dstM, eaM, ls, ld_, wed, elog, mx,
                                               (long long)Etot, 1);
  k_edge_expsum<<<nb(Etot, T), T, 0, stream>>>(dstM, elog, mx, den, (long long)Etot, 1);
  k_aggregate<<<nb(Etot, T), T, 0, stream>>>(srcM, dstM, elog, den, h3, out,
                                             (long long)Etot, 1, 1, 1);
  k_bias_act<<<nb(N, T), T, 0, stream>>>(out, b3, N, 1, 0);
}